// KCSetGNN_69028714381410
// MI455X (gfx1250) — compile-verified
//
#include <hip/hip_runtime.h>
#include <hip/hip_bf16.h>

typedef __attribute__((ext_vector_type(2))) float v2f;
typedef __attribute__((ext_vector_type(8))) float v8f;

#define N_NODES_C 80000
#define N_EDGES_C 800000
#define NSN_C     400000
#define NSE_C     800000
#define NSUB_C    102400
#define EBIP_C    500000
#define NGRAPH_C  128
#define NHID_C    128

// ---------------------------------------------------------------- utilities
__global__ void fill_f32(float* p, float v, long long n) {
    long long i = (long long)blockIdx.x * blockDim.x + threadIdx.x;
    if (i < n) p[i] = v;
}
__global__ void fill_u32(unsigned* p, unsigned v, long long n) {
    long long i = (long long)blockIdx.x * blockDim.x + threadIdx.x;
    if (i < n) p[i] = v;
}

__device__ __forceinline__ unsigned ford(float f) {
    unsigned u = __float_as_uint(f);
    return (u & 0x80000000u) ? ~u : (u | 0x80000000u);
}
__device__ __forceinline__ float ford_inv(unsigned u) {
    return __uint_as_float((u & 0x80000000u) ? (u & 0x7fffffffu) : ~u);
}

// ------------------------------------------------ WMMA f32 GEMM: C = A @ W
// A:[nrows,128] row-major, W:[128,128] row-major, C:[nrows,128].
// EPI: 0 = raw, 1 = relu(C + bias). In-place (C==A) is safe per-block.
// Block = 128 threads (4 waves); wave w handles rows [blk*64+w*16, +16).
// LDS (dynamic, 64KB): W transposed with XOR swizzle: wt[n*128 + (k ^ ((n&15)<<2))] = W[k*128+n]
template <int EPI>
__global__ __launch_bounds__(128) void gemm128_wmma(
    const float* __restrict__ A, const float* __restrict__ W,
    const float* __restrict__ bias, float* __restrict__ C, int nrows) {
    extern __shared__ float wt[];  // 128*128 floats = 64KB
    const int t = threadIdx.x;
    const int lane = t & 31;
    const int wave = t >> 5;

    // cooperative transpose of W into LDS (coalesced global reads)
    for (int idx = t; idx < 128 * 128; idx += 128) {
        int k = idx >> 7, n = idx & 127;
        wt[n * 128 + (k ^ ((n & 15) << 2))] = W[idx];
    }
    __syncthreads();

    const long long row0 = (long long)blockIdx.x * 64 + wave * 16;
    const int m  = lane & 15;        // row (A) / col (B) within 16-tile
    const int kh = (lane >> 4) * 2;  // K sub-offset: lanes 16-31 handle K+2,K+3
    const int sw = m << 2;           // LDS bank swizzle

    v8f acc[8] = {};
    const float* ap = A + (row0 + m) * 128 + kh;

    for (int k0 = 0; k0 < 128; k0 += 4) {
        v2f a;
        a.x = ap[k0];
        a.y = ap[k0 + 1];
        const float* bbase = wt + m * 128 + ((k0 + kh) ^ sw);
#pragma unroll
        for (int ct = 0; ct < 8; ct++) {
            const float* bp = bbase + ct * 16 * 128;
            v2f b;
            b.x = bp[0];
            b.y = bp[1];
            acc[ct] = __builtin_amdgcn_wmma_f32_16x16x4_f32(
                false, a, false, b, (short)0, acc[ct], false, false);
        }
    }

    // C/D layout: VGPR i -> row i (lanes 0-15) / row i+8 (lanes 16-31), col = lane&15
    const int rbase = (lane >> 4) * 8;
#pragma unroll
    for (int ct = 0; ct < 8; ct++) {
        int col = ct * 16 + m;
        float badd = (EPI == 1) ? bias[col] : 0.f;
#pragma unroll
        for (int i = 0; i < 8; i++) {
            float v = acc[ct][i];
            if (EPI == 1) v = fmaxf(v + badd, 0.f);
            C[(row0 + rbase + i) * 128 + col] = v;
        }
    }
}

// ------------------------------------------------ encoders
__global__ void node_encoder(const float* __restrict__ x, const float* __restrict__ W,
                             const float* __restrict__ b, float* __restrict__ h) {
    __shared__ float xs[16 * 16];
    int r0 = blockIdx.x * 16;
    int t = threadIdx.x;  // 128
    for (int i = t; i < 256; i += 128) xs[i] = x[(long long)r0 * 16 + i];
    __syncthreads();
    float w[16];
#pragma unroll
    for (int k = 0; k < 16; k++) w[k] = W[k * 128 + t];
    float bb = b[t];
    for (int r = 0; r < 16; r++) {
        float s = bb;
#pragma unroll
        for (int k = 0; k < 16; k++) s += xs[r * 16 + k] * w[k];
        h[(long long)(r0 + r) * 128 + t] = fmaxf(s, 0.f);
    }
}

__global__ void edge_encoder(const float* __restrict__ eattr, const float* __restrict__ pos,
                             const int* __restrict__ ei, const float* __restrict__ We,
                             const float* __restrict__ be, const float* __restrict__ Wp,
                             const float* __restrict__ bp, float* __restrict__ ea) {
    __shared__ float es[8 * 8];
    __shared__ float dsh[8 * 3];
    int e0 = blockIdx.x * 8;
    int t = threadIdx.x;  // 128
    if (t < 64) es[t] = eattr[(long long)e0 * 8 + t];
    if (t < 8) {
        int e = e0 + t;
        int s = ei[e], d = ei[N_EDGES_C + e];
        dsh[t * 3 + 0] = fabsf(pos[(long long)s * 3 + 0] - pos[(long long)d * 3 + 0]);
        dsh[t * 3 + 1] = fabsf(pos[(long long)s * 3 + 1] - pos[(long long)d * 3 + 1]);
        dsh[t * 3 + 2] = fabsf(pos[(long long)s * 3 + 2] - pos[(long long)d * 3 + 2]);
    }
    __syncthreads();
    float we[8], wp3[3];
#pragma unroll
    for (int k = 0; k < 8; k++) we[k] = We[k * 128 + t];
#pragma unroll
    for (int k = 0; k < 3; k++) wp3[k] = Wp[k * 128 + t];
    float bbe = be[t], bbp = bp[t];
    for (int r = 0; r < 8; r++) {
        float s1 = bbe;
#pragma unroll
        for (int k = 0; k < 8; k++) s1 += es[r * 8 + k] * we[k];
        float s2 = bbp;
#pragma unroll
        for (int k = 0; k < 3; k++) s2 += dsh[r * 3 + k] * wp3[k];
        ea[(long long)(e0 + r) * 128 + t] = fmaxf(s1, 0.f) + fmaxf(s2, 0.f);
    }
}

// ------------------------------------------------ gathers / scatters
__global__ void gather_rows(const float* __restrict__ src, const int* __restrict__ map,
                            float* __restrict__ dst, int n) {
    long long tid = (long long)blockIdx.x * blockDim.x + threadIdx.x;
    if (tid >= (long long)n * 32) return;
    int i = (int)(tid >> 5), j = (int)(tid & 31);
    const float4* s = (const float4*)(src + (long long)map[i] * 128);
    ((float4*)(dst + (long long)i * 128))[j] = s[j];
}

__global__ void conv_message(const float* __restrict__ sx, const float* __restrict__ ea,
                             const int* __restrict__ cs, const int* __restrict__ sem,
                             float* __restrict__ agg) {
    long long tid = (long long)blockIdx.x * blockDim.x + threadIdx.x;
    if (tid >= (long long)NSE_C * 128) return;
    int e = (int)(tid >> 7), j = (int)(tid & 127);
    int s = cs[e];
    int d = cs[NSE_C + e];
    float m = sx[(long long)s * 128 + j] + ea[(long long)sem[e] * 128 + j];
    atomicAdd(&agg[(long long)d * 128 + j], fmaxf(m, 0.f));
}

__global__ void col_stats(const float* __restrict__ T, float* __restrict__ sum,
                          float* __restrict__ sumsq, int nrows) {
    int j = threadIdx.x;  // 128
    long long r0 = (long long)blockIdx.x * 64;
    float s = 0.f, sq = 0.f;
    int rend = (nrows - (int)r0 < 64) ? (nrows - (int)r0) : 64;
    for (int r = 0; r < rend; r++) {
        float v = T[(r0 + r) * 128 + j];
        s += v;
        sq += v * v;
    }
    atomicAdd(&sum[j], s);
    atomicAdd(&sumsq[j], sq);
}

__global__ void bn_relu_res(const float* __restrict__ T, const float* __restrict__ sum,
                            const float* __restrict__ sumsq, float* __restrict__ sx,
                            long long nrows) {
    long long tid = (long long)blockIdx.x * blockDim.x + threadIdx.x;
    if (tid >= nrows * 128) return;
    int j = (int)(tid & 127);
    float inv_n = 1.f / (float)nrows;
    float mu = sum[j] * inv_n;
    float var = fmaxf(sumsq[j] * inv_n - mu * mu, 0.f);
    float inv = rsqrtf(var + 1e-5f);
    float v = (T[tid] - mu) * inv;
    sx[tid] += fmaxf(v, 0.f);
}

__global__ void pool_rows(const float* __restrict__ sx, const int* __restrict__ sb,
                          float* __restrict__ subs) {
    long long tid = (long long)blockIdx.x * blockDim.x + threadIdx.x;
    if (tid >= (long long)NSN_C * 128) return;
    int i = (int)(tid >> 7), j = (int)(tid & 127);
    atomicAdd(&subs[(long long)sb[i] * 128 + j], sx[tid]);
}

__global__ void subs_pre(float* __restrict__ subs, const int* __restrict__ ks,
                         const int* __restrict__ nc, const float* __restrict__ embn,
                         const float* __restrict__ embc) {
    long long tid = (long long)blockIdx.x * blockDim.x + threadIdx.x;
    if (tid >= (long long)NSUB_C * 128) return;
    int s = (int)(tid >> 7), j = (int)(tid & 127);
    int k = ks[s], c = nc[s];
    subs[tid] = subs[tid] / (float)(k + 1) + embn[k * 128 + j] + embc[c * 128 + j];
}

__global__ void bip_gather(const float* __restrict__ subs, const int* __restrict__ bip,
                           float* __restrict__ msgacc, float* __restrict__ cnt) {
    long long tid = (long long)blockIdx.x * blockDim.x + threadIdx.x;
    if (tid >= (long long)EBIP_C * 128) return;
    int e = (int)(tid >> 7), j = (int)(tid & 127);
    int s = bip[e];
    int d = bip[EBIP_C + e];
    atomicAdd(&msgacc[(long long)d * 128 + j], subs[(long long)s * 128 + j]);
    if (j == 0) atomicAdd(&cnt[d], 1.f);
}

__global__ void bip_combine(float* __restrict__ subs, const float* __restrict__ msg,
                            const float* __restrict__ cnt, const float* __restrict__ bbip) {
    long long tid = (long long)blockIdx.x * blockDim.x + threadIdx.x;
    if (tid >= (long long)NSUB_C * 128) return;
    int s = (int)(tid >> 7), j = (int)(tid & 127);
    subs[tid] = fmaxf(subs[tid] + msg[tid] + cnt[s] * bbip[j], 0.f);
}

// ------------------------------------------------ readout
__global__ void readout_scatter(const float* __restrict__ subs, const int* __restrict__ ks,
                                const int* __restrict__ nc, const int* __restrict__ bsub,
                                const float* __restrict__ embn, const float* __restrict__ embc,
                                const float* __restrict__ Wgate, float* __restrict__ suminfo,
                                unsigned* __restrict__ maxinfo) {
    __shared__ float kc[32];
    int s = blockIdx.x;
    int j = threadIdx.x;  // 128
    int k = ks[s];
    if (j < 16) kc[j] = embn[k * 16 + j];
    else if (j < 32) kc[j] = embc[nc[s] * 16 + (j - 16)];
    __syncthreads();
    float g = 0.f;
#pragma unroll
    for (int kk = 0; kk < 32; kk++) g += kc[kk] * Wgate[kk * 128 + j];
    g = 1.f / (1.f + expf(-g));
    int w = k + bsub[s] * 2;
    float v = subs[(long long)s * 128 + j];
    atomicAdd(&suminfo[w * 128 + j], v * g);
    atomicMax(&maxinfo[w * 128 + j], ford(v));
}

__global__ void dense0(const unsigned* __restrict__ maxinfo, const float* __restrict__ suminfo,
                       const float* __restrict__ Wd0, const float* __restrict__ bd0,
                       float* __restrict__ h2) {
    __shared__ float cat[256];
    int r = blockIdx.x, j = threadIdx.x;  // 256 x 128
    cat[j] = ford_inv(maxinfo[r * 128 + j]);
    cat[128 + j] = suminfo[r * 128 + j];
    __syncthreads();
    float s = bd0[j];
    for (int k = 0; k < 256; k++) s += cat[k] * Wd0[k * 128 + j];
    h2[r * 128 + j] = fmaxf(s, 0.f);
}

__global__ void dense1(const float* __restrict__ h2, const float* __restrict__ Wd1,
                       const float* __restrict__ bd1, float* __restrict__ t2) {
    __shared__ float row[256];
    int g = blockIdx.x, j = threadIdx.x;  // 128 x 128
    row[j] = h2[g * 256 + j];
    row[128 + j] = h2[g * 256 + 128 + j];
    __syncthreads();
    float s = bd1[j];
    for (int k = 0; k < 256; k++) s += row[k] * Wd1[k * 128 + j];
    t2[g * 128 + j] = fmaxf(s, 0.f);
}

__global__ void dense2(const float* __restrict__ t2, const float* __restrict__ Wd2,
                       const float* __restrict__ bd2, float* __restrict__ out) {
    int tid = blockIdx.x * blockDim.x + threadIdx.x;  // 2048
    if (tid >= NGRAPH_C * 16) return;
    int g = tid >> 4, o = tid & 15;
    float s = bd2[o];
    for (int j = 0; j < 128; j++) s += t2[g * 128 + j] * Wd2[j * 16 + o];
    out[g * 16 + o] = s;
}

// ---------------------------------------------------------------- launcher
extern "C" void kernel_launch(void* const* d_in, const int* in_sizes, int n_in,
                              void* d_out, int out_size, void* d_ws, size_t ws_size,
                              hipStream_t stream) {
    (void)in_sizes; (void)n_in; (void)out_size; (void)ws_size;
    const float* x         = (const float*)d_in[0];
    const float* edge_attr = (const float*)d_in[1];
    const float* pos       = (const float*)d_in[2];
    const int*   ei        = (const int*)d_in[3];
    const int*   snm       = (const int*)d_in[4];
    const int*   sem       = (const int*)d_in[5];
    const int*   cs        = (const int*)d_in[6];
    const int*   sb        = (const int*)d_in[7];
    const int*   ks        = (const int*)d_in[8];
    const int*   nc        = (const int*)d_in[9];
    const int*   bip       = (const int*)d_in[10];
    const int*   bsub      = (const int*)d_in[11];
    const float* W_in  = (const float*)d_in[13];
    const float* b_in  = (const float*)d_in[14];
    const float* W_edge= (const float*)d_in[15];
    const float* b_edge= (const float*)d_in[16];
    const float* W_pos = (const float*)d_in[17];
    const float* b_pos = (const float*)d_in[18];
    const float* Wc0   = (const float*)d_in[19];
    const float* Wc1   = (const float*)d_in[20];
    const float* Wbo   = (const float*)d_in[21];
    const float* bbo   = (const float*)d_in[22];
    const float* embns = (const float*)d_in[23];
    const float* embcs = (const float*)d_in[24];
    const float* Wso   = (const float*)d_in[25];
    const float* bso   = (const float*)d_in[26];
    const float* Wbip  = (const float*)d_in[27];
    const float* bbip  = (const float*)d_in[28];
    const float* embnk = (const float*)d_in[29];
    const float* embck = (const float*)d_in[30];
    const float* Wgate = (const float*)d_in[31];
    const float* Wd0   = (const float*)d_in[32];
    const float* bd0   = (const float*)d_in[33];
    const float* Wd1   = (const float*)d_in[34];
    const float* bd1   = (const float*)d_in[35];
    const float* Wd2   = (const float*)d_in[36];
    const float* bd2   = (const float*)d_in[37];

    // workspace carve (256B aligned)
    char* w = (char*)d_ws;
    auto carve = [&](size_t bytes) -> char* {
        char* p = w;
        w += (bytes + 255) & ~(size_t)255;
        return p;
    };
    float* h    = (float*)carve((size_t)N_NODES_C * 128 * 4);
    float* ea   = (float*)carve((size_t)N_EDGES_C * 128 * 4);
    float* sx   = (float*)carve((size_t)NSN_C * 128 * 4);
    float* agg  = (float*)carve((size_t)NSN_C * 128 * 4);
    float* subs = (float*)carve((size_t)NSUB_C * 128 * 4);
    float* msg  = (float*)carve((size_t)NSUB_C * 128 * 4);
    float* cnt  = (float*)carve((size_t)NSUB_C * 4);
    float* csum = (float*)carve(128 * 4);
    float* csq  = (float*)carve(128 * 4);
    unsigned* maxinfo = (unsigned*)carve(256 * 128 * 4);
    float* suminfo    = (float*)carve(256 * 128 * 4);
    float* h2   = (float*)carve(256 * 128 * 4);
    float* t2   = (float*)carve(128 * 128 * 4);

    const size_t LDSB = 128 * 128 * 4;  // 64KB for gemm128_wmma
    auto blks = [](long long n, int b) { return (unsigned)((n + b - 1) / b); };

    // encoders
    node_encoder<<<N_NODES_C / 16, 128, 0, stream>>>(x, W_in, b_in, h);
    edge_encoder<<<N_EDGES_C / 8, 128, 0, stream>>>(edge_attr, pos, ei, W_edge, b_edge,
                                                    W_pos, b_pos, ea);
    // gather subgraph node features
    gather_rows<<<blks((long long)NSN_C * 32, 256), 256, 0, stream>>>(h, snm, sx, NSN_C);

    // two GINE conv layers
    const float* Wc[2] = {Wc0, Wc1};
    for (int c = 0; c < 2; c++) {
        fill_f32<<<blks((long long)NSN_C * 128, 256), 256, 0, stream>>>(agg, 0.f,
                                                                        (long long)NSN_C * 128);
        conv_message<<<blks((long long)NSE_C * 128, 256), 256, 0, stream>>>(sx, ea, cs, sem, agg);
        gemm128_wmma<0><<<NSN_C / 64, 128, LDSB, stream>>>(agg, Wc[c], Wc[c], agg, NSN_C);
        fill_f32<<<1, 256, 0, stream>>>(csum, 0.f, 128);
        fill_f32<<<1, 256, 0, stream>>>(csq, 0.f, 128);
        col_stats<<<NSN_C / 64, 128, 0, stream>>>(agg, csum, csq, NSN_C);
        bn_relu_res<<<blks((long long)NSN_C * 128, 256), 256, 0, stream>>>(agg, csum, csq, sx,
                                                                           NSN_C);
    }
    // output encoder
    gemm128_wmma<1><<<NSN_C / 64, 128, LDSB, stream>>>(sx, Wbo, bbo, sx, NSN_C);

    // pool subgraphs
    fill_f32<<<blks((long long)NSUB_C * 128, 256), 256, 0, stream>>>(subs, 0.f,
                                                                     (long long)NSUB_C * 128);
    pool_rows<<<blks((long long)NSN_C * 128, 256), 256, 0, stream>>>(sx, sb, subs);
    subs_pre<<<blks((long long)NSUB_C * 128, 256), 256, 0, stream>>>(subs, ks, nc, embns, embcs);
    gemm128_wmma<1><<<NSUB_C / 64, 128, LDSB, stream>>>(subs, Wso, bso, subs, NSUB_C);

    // bipartite layer: msg = (segment_sum subs[bs]) @ Wbip + cnt*bbip
    fill_f32<<<blks((long long)NSUB_C * 128, 256), 256, 0, stream>>>(msg, 0.f,
                                                                     (long long)NSUB_C * 128);
    fill_f32<<<blks(NSUB_C, 256), 256, 0, stream>>>(cnt, 0.f, NSUB_C);
    bip_gather<<<blks((long long)EBIP_C * 128, 256), 256, 0, stream>>>(subs, bip, msg, cnt);
    gemm128_wmma<0><<<NSUB_C / 64, 128, LDSB, stream>>>(msg, Wbip, Wbip, msg, NSUB_C);
    bip_combine<<<blks((long long)NSUB_C * 128, 256), 256, 0, stream>>>(subs, msg, cnt, bbip);

    // gated readout
    fill_f32<<<blks(256 * 128, 256), 256, 0, stream>>>(suminfo, 0.f, 256 * 128);
    fill_u32<<<blks(256 * 128, 256), 256, 0, stream>>>(maxinfo, 0x007FFFFFu /*ord(-inf)*/,
                                                       256 * 128);
    readout_scatter<<<NSUB_C, 128, 0, stream>>>(subs, ks, nc, bsub, embnk, embck, Wgate,
                                                suminfo, maxinfo);
    // final MLP
    dense0<<<256, 128, 0, stream>>>(maxinfo, suminfo, Wd0, bd0, h2);
    dense1<<<NGRAPH_C, 128, 0, stream>>>(h2, Wd1, bd1, t2);
    dense2<<<blks(NGRAPH_C * 16, 256), 256, 0, stream>>>(t2, Wd2, bd2, (float*)d_out);
}